// InstrumentalModel_7370163880249
// MI455X (gfx1250) — compile-verified
//
#include <hip/hip_runtime.h>

// ---- MI455X / gfx1250, wave32 only ----
typedef __attribute__((ext_vector_type(2))) float v2f;
typedef __attribute__((ext_vector_type(8))) float v8f;

constexpr int NX     = 4000000;   // high_res_model length
constexpr int NSEG   = 500002;    // N_EDGES + 1
constexpr int NOUT   = 500000;
constexpr int TAPS   = 901;
constexpr int TILE   = 256;       // conv outputs per wave
constexpr int NTILES = NX / TILE; // 15625
constexpr int WPG    = 8;         // waves per workgroup
constexpr int XCOLS  = 84;        // staged window columns (stride; bank-conflict free)
constexpr int XWIN   = 16 * XCOLS;// 1344 staged signal values per tile
constexpr int GZLEN  = 1088;      // 64 front zeros + 901 taps + tail zeros

// Kernel 1: zero the segment-sum accumulator in workspace.
__global__ __launch_bounds__(256) void zero_seg(float* __restrict__ segsum) {
  int i = blockIdx.x * 256 + threadIdx.x;
  if (i < NSEG) segsum[i] = 0.0f;
}

// Kernel 2: Gaussian conv1d via V_WMMA_F32_16X16X4_F32 + segment-sum scatter.
// Tile of 256 outputs per wave: D[m][n] = out[i0 + 16m + n].
// Per (p, j):  A[m][k] = xs[p + 16*(m+k+4j)]          (Hankel signal slice, LDS)
//              B[k][n] = gz[64 + p - n - 49 + 16(k+4j) ... ] reversed gaussian window
// => each (m,n,k) product is g[t] * x[i + t - 450] with t = p - n - 49 + 16(k+4j);
//    sweeping p,j covers every tap exactly once per output (zeros pad the edges).
__global__ __launch_bounds__(256) void conv_wmma(
    const float* __restrict__ x,
    const float* __restrict__ ln_sigma,
    const float* __restrict__ kgrid,
    const int*   __restrict__ labels,
    float*       __restrict__ segsum)
{
  __shared__ float gz[GZLEN];
  __shared__ float xs[WPG][XWIN];

  const int tid  = threadIdx.x;
  const int wave = tid >> 5;
  const int lane = tid & 31;
  const int q    = lane & 15;
  const int hi   = lane >> 4;

  // --- stage zero-padded gaussian (whole block) ---
  {
    const float sig  = 0.01f + __expf(ln_sigma[0]);
    const float amp  = 0.01f / (sig * __builtin_sqrtf(6.2831853308f)); // TWO_PI as written
    const float c2   = 0.5f / (sig * sig);
    for (int t = tid; t < GZLEN; t += 256) {
      int g = t - 64;                       // 64 zeros in front
      float v = 0.0f;
      if (g >= 0 && g < TAPS) {
        float kg = kgrid[g];
        v = amp * __expf(-kg * kg * c2);
      }
      gz[t] = v;
    }
  }

  // --- stage this wave's signal window, phase-transposed ---
  const int tile   = blockIdx.x * WPG + wave;
  const bool active = (tile < NTILES);
  const int i0     = tile * TILE;
  const int wbase  = i0 - 499;              // i0 - 435 - 64 (extra 64 for low-tap phase)
  float* xw = &xs[wave][0];
  if (active) {
    for (int u = lane; u < XWIN; u += 32) {
      int gi = wbase + u;
      float v = (gi >= 0 && gi < NX) ? x[gi] : 0.0f;
      xw[(u & 15) * XCOLS + (u >> 4)] = v;  // row = phase, conflict-free (stride 84)
    }
  }
  __syncthreads();
  if (!active) return;

  // --- 256 WMMAs accumulate the full 901-tap conv for 256 outputs ---
  v8f acc = {0.f, 0.f, 0.f, 0.f, 0.f, 0.f, 0.f, 0.f};
  const int ab = q + 2 * hi;                // A col base:  m=q, k=2*hi / 2*hi+1
  for (int p = 0; p < 16; ++p) {
    const float* xrow = &xw[p * XCOLS];
    const float* grow = &gz[p + 15 - q + 32 * hi];
#pragma unroll
    for (int j = 0; j < 16; ++j) {
      v2f a, b;
      a.x = xrow[ab + 4 * j];               // A[m][2hi]   = h[m + k + 4j]
      a.y = xrow[ab + 4 * j + 1];           // A[m][2hi+1]
      b.x = grow[64 * j];                   // B[2hi][n]   = gz[... - n]
      b.y = grow[64 * j + 16];              // B[2hi+1][n]
      acc = __builtin_amdgcn_wmma_f32_16x16x4_f32(
          false, a, false, b, (short)0, acc, false, false);
    }
  }

  // --- scatter: D vgpr v, lane (q,hi) holds out[i0 + 16*(v+8*hi) + q] ---
#pragma unroll
  for (int v = 0; v < 8; ++v) {
    int oi  = i0 + 16 * (v + 8 * hi) + q;
    int lab = labels[oi];
    atomicAdd(&segsum[lab], acc[v]);
  }
}

// Kernel 3: seg mean -> clip -> * (design @ w + bias)
__global__ __launch_bounds__(256) void finalize(
    const float* __restrict__ segsum,
    const float* __restrict__ counts,
    const float* __restrict__ design,
    const float* __restrict__ w,
    const float* __restrict__ bias,
    float*       __restrict__ out)
{
  int o = blockIdx.x * 256 + threadIdx.x;
  if (o >= NOUT) return;
  int s = o + 1;                             // drop underflow bin
  float mean = segsum[s] / counts[s];
  mean = fminf(fmaxf(mean, 0.0f), 1.0f);
  float c = bias[0];
  const float* row = design + (long)o * 15;
#pragma unroll
  for (int p = 0; p < 15; ++p) c = fmaf(row[p], w[p], c);
  out[o] = mean * c;
}

extern "C" void kernel_launch(void* const* d_in, const int* in_sizes, int n_in,
                              void* d_out, int out_size, void* d_ws, size_t ws_size,
                              hipStream_t stream) {
  const float* x        = (const float*)d_in[0]; // high_res_model [4000000]
  const float* ln_sigma = (const float*)d_in[1]; // scalar
  const float* weight   = (const float*)d_in[2]; // [1,15]
  const float* bias     = (const float*)d_in[3]; // [1]
  const float* kgrid    = (const float*)d_in[4]; // [901]
  const float* design   = (const float*)d_in[5]; // [500000,15]
  const int*   labels   = (const int*)d_in[6];   // [4000000]
  const float* counts   = (const float*)d_in[7]; // [500002]
  float* out    = (float*)d_out;
  float* segsum = (float*)d_ws;                  // 500002 floats (2 MB scratch)

  zero_seg<<<(NSEG + 255) / 256, 256, 0, stream>>>(segsum);
  conv_wmma<<<(NTILES + WPG - 1) / WPG, 256, 0, stream>>>(x, ln_sigma, kgrid, labels, segsum);
  finalize<<<(NOUT + 255) / 256, 256, 0, stream>>>(segsum, counts, design, weight, bias, out);
}